// Grid2Graph_34815004901543
// MI455X (gfx1250) — compile-verified
//
#include <hip/hip_runtime.h>
#include <hip/hip_bf16.h>
#include <math.h>

#define B_  4
#define T_  8
#define C_  16
#define H_  64
#define W_  64
#define N_  2048
#define F_  128
#define P_  256
#define M_  2304   // N_ + P_
#define K_  10
#define KC_ 256    // C_*4*4 (im2col depth)

// GEMM tiling
#define BM 64
#define BN 32
#define BK 32
#define SKP (BK + 2)   // padded LDS row: keeps ds_load_b64 8B-aligned & bank-conflict-free

typedef float v2f __attribute__((ext_vector_type(2)));
typedef float v4f __attribute__((ext_vector_type(4)));
typedef float v8f __attribute__((ext_vector_type(8)));

// ---------------------------------------------------------------- utilities
__global__ void zero_i32(int* __restrict__ p, int n) {
  int g = blockIdx.x * blockDim.x + threadIdx.x;
  if (g < n) p[g] = 0;
}

// ------------------------------------------------------------ graph build
__global__ void build_loc(const float* __restrict__ latlon, float* __restrict__ loc) {
  int g = blockIdx.x * blockDim.x + threadIdx.x;      // B_*M_ threads, exact
  int b = g / M_, m = g - b * M_;
  float x, y;
  if (m < N_) {
    float lat = latlon[(b * N_ + m) * 2 + 0];
    float lon = latlon[(b * N_ + m) * 2 + 1];
    x = (lat + 90.0f) * (1.0f / 180.0f);
    y = (lon + 180.0f) * (1.0f / 360.0f);
  } else {
    int p = m - N_;
    x = (float)(p >> 4) * (1.0f / 15.0f);   // linspace(0,1,16)
    y = (float)(p & 15) * (1.0f / 15.0f);
  }
  loc[g * 2 + 0] = x;
  loc[g * 2 + 1] = y;
}

// one thread per row i: scan all M points (staged in LDS), keep 10 nearest.
// strict '<' while scanning ascending j reproduces top_k's lowest-index tiebreak.
__global__ void knn_kernel(const float* __restrict__ loc,
                           int* __restrict__ dstIdx, float* __restrict__ weArr,
                           float* __restrict__ deg, int* __restrict__ cnt) {
  __shared__ float sx[M_];
  __shared__ float sy[M_];
  int b = blockIdx.x / (M_ / 256);
  int i = (blockIdx.x % (M_ / 256)) * 256 + threadIdx.x;
  const float* lb = loc + (size_t)b * M_ * 2;
  for (int j = threadIdx.x; j < M_; j += 256) {
    sx[j] = lb[2 * j];
    sy[j] = lb[2 * j + 1];
  }
  __syncthreads();
  float px = sx[i], py = sy[i];
  float bd[K_];
  int   bi[K_];
#pragma unroll
  for (int k = 0; k < K_; ++k) { bd[k] = 3.0e38f; bi[k] = 0; }
  for (int j = 0; j < M_; ++j) {
    float dx = px - sx[j], dy = py - sy[j];
    float d2 = dx * dx + dy * dy;
    if (d2 < bd[K_ - 1]) {
      bd[K_ - 1] = d2; bi[K_ - 1] = j;
#pragma unroll
      for (int k = K_ - 1; k > 0; --k) {
        if (bd[k] < bd[k - 1]) {
          float td = bd[k]; bd[k] = bd[k - 1]; bd[k - 1] = td;
          int   ti = bi[k]; bi[k] = bi[k - 1]; bi[k - 1] = ti;
        }
      }
    }
  }
  int base = (b * M_ + i) * K_;
#pragma unroll
  for (int k = 0; k < K_; ++k) {
    float dist = sqrtf(bd[k] + 1e-24f);      // SAFE inside sqrt, as reference
    float w = 1.0f / (dist + 1e-10f);        // EPS_W
    dstIdx[base + k] = bi[k];
    weArr[base + k]  = w;
    atomicAdd(&deg[b * M_ + bi[k]], w);
    atomicAdd(&cnt[b * M_ + bi[k]], 1);
  }
}

// serial exclusive scan per batch (2304 elems; runs once)
__global__ void scan_kernel(const int* __restrict__ cnt, int* __restrict__ roff) {
  if (threadIdx.x != 0) return;
  int b = blockIdx.x;
  int acc = 0;
  for (int m = 0; m < M_; ++m) {
    roff[b * (M_ + 1) + m] = acc;
    acc += cnt[b * M_ + m];
  }
  roff[b * (M_ + 1) + M_] = acc;
}

__global__ void fill_kernel(const int* __restrict__ dstIdx, const float* __restrict__ weArr,
                            const int* __restrict__ roff, int* __restrict__ fillc,
                            int* __restrict__ rsrc, float* __restrict__ rw) {
  int g = blockIdx.x * blockDim.x + threadIdx.x;     // B_*M_*K_ threads, exact
  int b = g / (M_ * K_);
  int r = g - b * (M_ * K_);
  int i = r / K_;                                    // src node
  int d = dstIdx[g];
  int pos = roff[b * (M_ + 1) + d] + atomicAdd(&fillc[b * M_ + d], 1);
  rsrc[(size_t)b * M_ * K_ + pos] = i;
  rw  [(size_t)b * M_ * K_ + pos] = weArr[g];
}

// ------------------------------------------------------------ feature build
__global__ void gfeat_kernel(const float* __restrict__ gd, const float* __restrict__ nw,
                             const float* __restrict__ nb, float* __restrict__ h0) {
  int g = blockIdx.x * blockDim.x + threadIdx.x;     // N_*F_, exact
  int n = g >> 7, f = g & (F_ - 1);
  h0[g] = fmaf(gd[n], nw[f], nb[f]);
}

__global__ void im2col_kernel(const float* __restrict__ x, float* __restrict__ buf) {
  int g = blockIdx.x * blockDim.x + threadIdx.x;     // P_*KC_, exact
  int p  = g >> 8;
  int kc = g & 255;
  int ch = kc >> 4;
  int ky = (kc >> 2) & 3;
  int kx = kc & 3;
  int pr = p >> 4, pc = p & 15;
  buf[g] = x[(ch * H_ + (pr * 4 + ky)) * W_ + (pc * 4 + kx)];
}

// ------------------------------------------------------------ aggregation
// agg[m,f] = sum_e rw[e] * h[rsrc[e], f] over reverse-CSR row m
__global__ void aggregate_kernel(const float* __restrict__ h, int Fd,
                                 const int* __restrict__ roff, const int* __restrict__ rsrc,
                                 const float* __restrict__ rw, float* __restrict__ agg) {
  int g = blockIdx.x * blockDim.x + threadIdx.x;     // Mrows*Fd, exact
  int m = g / Fd;
  int f = g - m * Fd;
  float s = 0.0f;
  int e1 = roff[m + 1];
  for (int e = roff[m]; e < e1; ++e)
    s = fmaf(rw[e], h[(size_t)rsrc[e] * Fd + f], s);
  agg[g] = s;
}

// ------------------------------------------------------------ WMMA f32 GEMM
// C[m,j] = sum_k A[m,k]*W[k*wsK + j*wsJ] + scale[m]*bias[j]  (scale==null -> 1)
// Block 256 thr = 8 wave32s; block tile 64x32, each wave one 16x16 tile.
// Double-buffered LDS staging: A (64xBK) row-major, W tile transposed (32xBK)
// so both fragment reads are ds_load_b64. Pad SKP=BK+2 -> lanes hit disjoint
// bank quads (34*r mod 64 are >=4 apart). Grids divide exactly -> EXEC all-1s
// at every WMMA (ISA 7.12 requirement). Fragment layouts per ISA 7.12.2.
__global__ void gemm_wmma(const float* __restrict__ A, int lda,
                          const float* __restrict__ Wm, int wsK, int wsJ,
                          const float* __restrict__ scale,
                          const float* __restrict__ bias,
                          float* __restrict__ Cout, int ldc,
                          int Kdim, int relu) {
  __shared__ __align__(16) float sA[2][BM * SKP];
  __shared__ __align__(16) float sB[2][BN * SKP];

  int tid  = threadIdx.x;
  int wave = tid >> 5;
  int lane = tid & 31;
  int half = lane >> 4;          // 0/1 -> K pair (ISA: lanes 16-31 hold K+2)
  int lrow = lane & 15;
  int waveM = wave >> 1;         // 0..3
  int waveN = wave & 1;          // 0..1
  int blockM = blockIdx.x * BM;
  int blockN = blockIdx.y * BN;

  // coalesced global->reg staging mappings
  int arow = tid >> 2;           // 0..63
  int acol = (tid & 3) * 8;      // 0,8,16,24
  const float* Aload = A + (size_t)(blockM + arow) * lda + acol;

  int bj  = tid & 31;            // column j within tile (fast index -> coalesced for wsJ==1)
  int bkq = (tid >> 5) * 4;      // k quad 0,4,...,28
  const float* Bload = Wm + (size_t)(blockN + bj) * wsJ + (size_t)bkq * wsK;

  v4f a0, a1;
  float br[4];
  v8f acc = {};

  auto loadTile = [&](int k0) {
    const float* ap = Aload + k0;
    a0 = *(const v4f*)(ap);
    a1 = *(const v4f*)(ap + 4);
    const float* bp = Bload + (size_t)k0 * wsK;
    br[0] = bp[0];
    br[1] = bp[(size_t)wsK];
    br[2] = bp[(size_t)2 * wsK];
    br[3] = bp[(size_t)3 * wsK];
  };
  auto storeTile = [&](int buf) {
    float* sa = &sA[buf][arow * SKP + acol];
    sa[0] = a0.x; sa[1] = a0.y; sa[2] = a0.z; sa[3] = a0.w;
    sa[4] = a1.x; sa[5] = a1.y; sa[6] = a1.z; sa[7] = a1.w;
    float* sb = &sB[buf][bj * SKP + bkq];
    sb[0] = br[0]; sb[1] = br[1]; sb[2] = br[2]; sb[3] = br[3];
  };
  auto compute = [&](int buf) {
    const float* pa = &sA[buf][(waveM * 16 + lrow) * SKP + half * 2];
    const float* pb = &sB[buf][(waveN * 16 + lrow) * SKP + half * 2];
#pragma unroll
    for (int kk = 0; kk < BK; kk += 4) {
      v2f af = *(const v2f*)(pa + kk);
      v2f bf = *(const v2f*)(pb + kk);
      acc = __builtin_amdgcn_wmma_f32_16x16x4_f32(false, af, false, bf,
                                                  (short)0, acc, false, false);
    }
  };

  int nk = Kdim / BK;
  loadTile(0);
  storeTile(0);
  __syncthreads();
  for (int kt = 0; kt < nk; ++kt) {
    if (kt + 1 < nk) loadTile((kt + 1) * BK);   // prefetch next tile into regs
    compute(kt & 1);
    if (kt + 1 < nk) {
      __syncthreads();
      storeTile((kt + 1) & 1);
      __syncthreads();
    }
  }

  // epilogue: D layout -> VGPR r: M = tileM + r (+8 for lanes 16-31), N = lane&15
  int tileM = blockM + waveM * 16;
  int ccol  = blockN + waveN * 16 + lrow;
  float bjv = bias[ccol];
  int mbase = tileM + half * 8;
#pragma unroll
  for (int r = 0; r < 8; ++r) {
    int m = mbase + r;
    float s = scale ? scale[m] : 1.0f;
    float v = acc[r] + s * bjv;
    if (relu) v = fmaxf(v, 0.0f);
    Cout[(size_t)m * ldc + ccol] = v;
  }
}

// ------------------------------------------------------------ LayerNorm (F=128)
__global__ void layernorm_kernel(const float* __restrict__ x, const float* __restrict__ g,
                                 const float* __restrict__ bb, float* __restrict__ out) {
  __shared__ float red[F_];
  int n = blockIdx.x, f = threadIdx.x;
  float v = x[(size_t)n * F_ + f];
  red[f] = v;
  __syncthreads();
  for (int s = F_ / 2; s > 0; s >>= 1) {
    if (f < s) red[f] += red[f + s];
    __syncthreads();
  }
  float mu = red[0] * (1.0f / F_);
  __syncthreads();
  float d = v - mu;
  red[f] = d * d;
  __syncthreads();
  for (int s = F_ / 2; s > 0; s >>= 1) {
    if (f < s) red[f] += red[f + s];
    __syncthreads();
  }
  float var = red[0] * (1.0f / F_);
  out[(size_t)n * F_ + f] = d * (1.0f / sqrtf(var + 1e-5f)) * g[f] + bb[f];
}

// ---------------------------------------------------------------- launcher
extern "C" void kernel_launch(void* const* d_in, const int* in_sizes, int n_in,
                              void* d_out, int out_size, void* d_ws, size_t ws_size,
                              hipStream_t stream) {
  const float* grid_data  = (const float*)d_in[0];
  const float* graph_data = (const float*)d_in[1];
  const float* latlon     = (const float*)d_in[2];
  // d_in[3], d_in[4]: time indices -- unused by reference math
  const float* conv_w = (const float*)d_in[5];
  const float* conv_b = (const float*)d_in[6];
  const float* ne_w   = (const float*)d_in[7];
  const float* ne_b   = (const float*)d_in[8];
  const float* W1     = (const float*)d_in[9];
  const float* b1     = (const float*)d_in[10];
  const float* W2     = (const float*)d_in[11];
  const float* b2     = (const float*)d_in[12];
  const float* ln_g   = (const float*)d_in[13];
  const float* ln_b   = (const float*)d_in[14];
  float* out = (float*)d_out;
  (void)in_sizes; (void)n_in; (void)out_size; (void)ws_size;

  // workspace carve-up (~10 MB total), 256B aligned
  size_t off = 0;
  char* base = (char*)d_ws;
  auto alloc = [&](size_t bytes) -> char* {
    char* p = base + off;
    off += (bytes + 255) & ~(size_t)255;
    return p;
  };
  float* loc    = (float*)alloc((size_t)B_ * M_ * 2 * 4);
  int*   dstIdx = (int*)  alloc((size_t)B_ * M_ * K_ * 4);
  float* weArr  = (float*)alloc((size_t)B_ * M_ * K_ * 4);
  float* deg    = (float*)alloc((size_t)B_ * M_ * 4);
  int*   cnt    = (int*)  alloc((size_t)B_ * M_ * 4);
  int*   fillc  = (int*)  alloc((size_t)B_ * M_ * 4);
  int*   roff   = (int*)  alloc((size_t)B_ * (M_ + 1) * 4);
  int*   rsrc   = (int*)  alloc((size_t)B_ * M_ * K_ * 4);
  float* rw     = (float*)alloc((size_t)B_ * M_ * K_ * 4);
  float* imbuf  = (float*)alloc((size_t)P_ * KC_ * 4);
  float* h0     = (float*)alloc((size_t)M_ * F_ * 4);
  float* agg1   = (float*)alloc((size_t)M_ * F_ * 4);
  float* h1     = (float*)alloc((size_t)M_ * 2 * F_ * 4);
  float* agg2   = (float*)alloc((size_t)N_ * 2 * F_ * 4);
  float* out2   = (float*)alloc((size_t)N_ * F_ * 4);

  // ---- per-batch KNN graph (independent of t) ----
  zero_i32<<<(B_ * M_ + 255) / 256, 256, 0, stream>>>((int*)deg, B_ * M_);
  zero_i32<<<(B_ * M_ + 255) / 256, 256, 0, stream>>>(cnt, B_ * M_);
  zero_i32<<<(B_ * M_ + 255) / 256, 256, 0, stream>>>(fillc, B_ * M_);
  build_loc<<<(B_ * M_) / 256, 256, 0, stream>>>(latlon, loc);
  knn_kernel<<<B_ * (M_ / 256), 256, 0, stream>>>(loc, dstIdx, weArr, deg, cnt);
  scan_kernel<<<B_, 32, 0, stream>>>(cnt, roff);
  fill_kernel<<<(B_ * M_ * K_) / 256, 256, 0, stream>>>(dstIdx, weArr, roff, fillc, rsrc, rw);

  // ---- per (b,t): features -> GNN -> LayerNorm (stream-ordered buffer reuse) ----
  for (int b = 0; b < B_; ++b) {
    const int*   roff_b = roff + (size_t)b * (M_ + 1);
    const int*   rsrc_b = rsrc + (size_t)b * M_ * K_;
    const float* rw_b   = rw   + (size_t)b * M_ * K_;
    const float* deg_b  = deg  + (size_t)b * M_;
    for (int t = 0; t < T_; ++t) {
      const float* gd = graph_data + (size_t)(b * T_ + t) * N_;
      const float* gx = grid_data  + (size_t)(b * T_ + t) * C_ * H_ * W_;

      gfeat_kernel<<<(N_ * F_) / 256, 256, 0, stream>>>(gd, ne_w, ne_b, h0);
      im2col_kernel<<<(P_ * KC_) / 256, 256, 0, stream>>>(gx, imbuf);
      // conv-as-GEMM: (256 x 256) x (256 x 128); W(k,j) = conv_w[j*256 + k]
      gemm_wmma<<<dim3(P_ / BM, F_ / BN), 256, 0, stream>>>(
          imbuf, KC_, conv_w, /*wsK=*/1, /*wsJ=*/KC_,
          nullptr, conv_b, h0 + (size_t)N_ * F_, F_, KC_, 0);

      // layer 1: agg(h0) @ W1 + deg*b1, ReLU  -> h1 (M x 256)
      aggregate_kernel<<<(M_ * F_) / 256, 256, 0, stream>>>(h0, F_, roff_b, rsrc_b, rw_b, agg1);
      gemm_wmma<<<dim3(M_ / BM, (2 * F_) / BN), 256, 0, stream>>>(
          agg1, F_, W1, /*wsK=*/2 * F_, /*wsJ=*/1,
          deg_b, b1, h1, 2 * F_, F_, 1);

      // layer 2 (only first N rows needed): agg(h1) @ W2 + deg*b2 -> out2 (N x 128)
      aggregate_kernel<<<(N_ * 2 * F_) / 256, 256, 0, stream>>>(h1, 2 * F_, roff_b, rsrc_b, rw_b, agg2);
      gemm_wmma<<<dim3(N_ / BM, F_ / BN), 256, 0, stream>>>(
          agg2, 2 * F_, W2, /*wsK=*/F_, /*wsJ=*/1,
          deg_b, b2, out2, F_, 2 * F_, 0);

      layernorm_kernel<<<N_, F_, 0, stream>>>(out2, ln_g, ln_b,
                                              out + (size_t)(b * T_ + t) * N_ * F_);
    }
  }
}